// WeightPooling_40415642255851
// MI455X (gfx1250) — compile-verified
//
#include <hip/hip_runtime.h>
#include <stdint.h>

// Problem constants (match reference): B=32, C=512, H=W=56, K=64
#define BB 32
#define CC 512
#define HW 3136          // 56*56
#define KK 64
#define NTHREADS 256

// LDS byte offset of a __shared__ object: the generic (flat) address of an LDS
// object is {shared_aperture[63:32], lds_offset[31:0]}, so truncation gives the
// wave-relative LDS byte address the async-to-LDS instructions expect in VDST.
__device__ __forceinline__ unsigned lds_off_of(const void* p) {
  return (unsigned)(uintptr_t)p;
}

__device__ __forceinline__ void async_copy_b128(unsigned lds_dst, const void* gsrc) {
  // GLOBAL_LOAD_ASYNC_TO_LDS_B128, GV mode (64-bit per-lane address, SADDR=off).
  // Tracked by ASYNCcnt, data never passes through VGPRs.
  asm volatile("global_load_async_to_lds_b128 %0, %1, off"
               :: "v"(lds_dst), "v"((unsigned long long)(uintptr_t)gsrc)
               : "memory");
}

__device__ __forceinline__ void wait_async0() {
#if __has_builtin(__builtin_amdgcn_s_wait_asynccnt)
  __builtin_amdgcn_s_wait_asynccnt(0);
#else
  asm volatile("s_wait_asynccnt 0" ::: "memory");
#endif
}

__global__ __launch_bounds__(NTHREADS)
void topk_weightpool_kernel(const float* __restrict__ x,
                            const float* __restrict__ att,
                            const float* __restrict__ w,
                            const float* __restrict__ bias,
                            float* __restrict__ out,
                            float* __restrict__ locs) {
  __shared__ float xs[HW];                         // 12544 B
  __shared__ float as[HW];                         // 12544 B
  __shared__ unsigned long long keys[HW];          // 25088 B
  __shared__ unsigned long long sel64[KK];         // 512 B
  __shared__ float partial[KK];                    // 256 B
  __shared__ unsigned hist[16];
  __shared__ unsigned sel_s, krem_s, cnt_s;

  const int row = blockIdx.x;            // row = b*C + c  (x is [B,C,H*W] row-major)
  const int c   = row & (CC - 1);
  const int tid = threadIdx.x;

  const float* xrow = x   + (size_t)row * HW;
  const float* arow = att + (size_t)c   * HW;

  // ---- Stage 1: async DMA both rows into LDS (784 float4 chunks each) ----
  const unsigned xs_base = lds_off_of(xs);
  const unsigned as_base = lds_off_of(as);
  for (int j = tid; j < 2 * (HW / 4); j += NTHREADS) {
    const bool isx = j < (HW / 4);
    const int  cj  = isx ? j : j - (HW / 4);
    async_copy_b128((isx ? xs_base : as_base) + cj * 16,
                    (isx ? xrow : arow) + cj * 4);
  }
  wait_async0();
  __syncthreads();

  // ---- Stage 2: build unique 64-bit sort keys ----
  // monotone map: u = bits ^ ((bits>>31) ? 0xFFFFFFFF : 0x80000000)  (u asc <=> val asc)
  // key = (~u)<<32 | idx  => ascending key == descending value, ties -> lower idx first
  for (int i = tid; i < HW; i += NTHREADS) {
    const float v = xs[i] * as[i];
    const unsigned bits = __float_as_uint(v);
    const unsigned u = bits ^ ((unsigned)((int)bits >> 31) | 0x80000000u);
    keys[i] = ((unsigned long long)(~u) << 32) | (unsigned)i;
  }
  __syncthreads();

  // ---- Stage 3: exact radix select of the rank-63 (0-based) smallest key ----
  unsigned long long prefix = 0ull;
  unsigned krank = KK - 1;
  for (int d = 15; d >= 0; --d) {
    const int sh = d * 4;
    if (tid < 16) hist[tid] = 0;
    __syncthreads();
    for (int i = tid; i < HW; i += NTHREADS) {
      const unsigned long long key = keys[i];
      const bool match = (d == 15) || ((key >> (sh + 4)) == prefix);
      if (match) atomicAdd(&hist[(unsigned)(key >> sh) & 15u], 1u);
    }
    __syncthreads();
    if (tid == 0) {
      unsigned cum = 0, b2 = 0;
      for (; b2 < 16; ++b2) {
        const unsigned h = hist[b2];
        if (krank < cum + h) break;
        cum += h;
      }
      sel_s  = b2;
      krem_s = krank - cum;
    }
    __syncthreads();
    prefix = (prefix << 4) | sel_s;
    krank  = krem_s;
  }
  const unsigned long long kth = prefix;   // exact 64th-smallest key (keys unique)

  // ---- Stage 4: compact the exactly-64 keys <= kth ----
  if (tid == 0) cnt_s = 0;
  __syncthreads();
  for (int i = tid; i < HW; i += NTHREADS) {
    const unsigned long long key = keys[i];
    if (key <= kth) {
      const unsigned p = atomicAdd(&cnt_s, 1u);
      if (p < KK) sel64[p] = key;
    }
  }

  // ---- Stage 5: bitonic sort 64 keys ascending (descending values) ----
  for (unsigned size2 = 2; size2 <= KK; size2 <<= 1) {
    for (unsigned stride = size2 >> 1; stride > 0; stride >>= 1) {
      __syncthreads();
      if (tid < KK) {
        const unsigned partner = (unsigned)tid ^ stride;
        if (partner > (unsigned)tid) {
          const unsigned long long a = sel64[tid];
          const unsigned long long b = sel64[partner];
          const bool asc = ((tid & size2) == 0);
          if (asc ? (a > b) : (a < b)) { sel64[tid] = b; sel64[partner] = a; }
        }
      }
    }
  }
  __syncthreads();

  // ---- Stage 6: emit locs, weighted reduce + bias ----
  if (tid < KK) {
    const unsigned long long key = sel64[tid];
    const unsigned idx = (unsigned)(key & 0xFFFFFFFFu);
    const unsigned u   = ~(unsigned)(key >> 32);
    const unsigned bits = (u & 0x80000000u) ? (u ^ 0x80000000u) : ~u;  // exact inverse map
    const float val = __uint_as_float(bits);
    locs[(size_t)row * KK + tid] = (float)idx;
    partial[tid] = val * w[c * KK + tid];
  }
  __syncthreads();
  if (tid == 0) {
    float s = bias[c];
#pragma unroll
    for (int i = 0; i < KK; ++i) s += partial[i];
    out[row] = s;
  }
}

extern "C" void kernel_launch(void* const* d_in, const int* in_sizes, int n_in,
                              void* d_out, int out_size, void* d_ws, size_t ws_size,
                              hipStream_t stream) {
  const float* x    = (const float*)d_in[0];  // [B,C,H,W]
  const float* att  = (const float*)d_in[1];  // [1,C,H,W]
  const float* w    = (const float*)d_in[2];  // [C,K]
  const float* bias = (const float*)d_in[3];  // [C]
  float* out  = (float*)d_out;                // [B,C]
  float* locs = out + (size_t)BB * CC;        // [B,C,K] as float
  topk_weightpool_kernel<<<dim3(BB * CC), dim3(NTHREADS), 0, stream>>>(
      x, att, w, bias, out, locs);
}